// HeteroRGCNLayer_20830591385607
// MI455X (gfx1250) — compile-verified
//
#include <hip/hip_runtime.h>
#include <hip/hip_bf16.h>

typedef __attribute__((ext_vector_type(2))) float v2f;
typedef __attribute__((ext_vector_type(8))) float v8f;

#define NPAPER   50000
#define NAUTHOR  30000
#define NSUBJECT 60
#define DIM      128

// ---------------------------------------------------------------------------
// Zero-fill (out buffer + degree arrays); harness poisons with 0xAA.
// ---------------------------------------------------------------------------
__global__ void fill_zero_kernel(float* __restrict__ p, long long n) {
    long long i = (long long)blockIdx.x * blockDim.x + threadIdx.x;
    if (i < n) p[i] = 0.0f;
}

// ---------------------------------------------------------------------------
// Y[N,128] = X[N,128] @ W[128,128] + b  using V_WMMA_F32_16X16X4_F32.
// Block = 256 threads = 8 waves. Block b covers rows [16b,16b+16);
// wave w covers cols [16w,16w+16). K loop: 128 in steps of 4 (32 WMMAs).
//
// ISA VGPR layouts (CDNA5 ISA, 32-bit matrices, wave32):
//   A 16x4 : lane(0-15)=M=r, VGPRv holds K = k + v + 2*half
//   B 4x16 : lane(0-15)=N=r, VGPRv holds K = k + v + 2*half (lanes16-31: K+2)
//   C/D    : VGPR j -> M = j + 8*half, N = r
//
// Out-of-range A rows are CLAMPED, not zeroed: row m of A only feeds row m
// of D, and the store guard drops rows >= N. This keeps control flow uniform
// (EXEC all-ones for WMMA) and the hot loop branch-free.
// ---------------------------------------------------------------------------
__global__ __launch_bounds__(256) void gemm_bias_wmma_kernel(
    const float* __restrict__ X, const float* __restrict__ W,
    const float* __restrict__ bias, float* __restrict__ Y, int N) {
    const int wave = threadIdx.x >> 5;
    const int lane = threadIdx.x & 31;
    const int half = lane >> 4;          // 0: lanes 0-15, 1: lanes 16-31
    const int r    = lane & 15;
    const int row_base = blockIdx.x << 4;
    const int col      = (wave << 4) + r;
    const int arow     = min(row_base + r, N - 1);   // clamp, see note above

    // Per-lane bases: all K-loop accesses become immediate offsets off these.
    const float* xptr = X + (long long)arow * DIM + 2 * half;          // A: +k*4B
    const float* wptr = W + (long long)(2 * half) * DIM + col;         // B: +k*512B

    v8f acc = {};
#pragma unroll
    for (int k = 0; k < DIM; k += 4) {
        const float2 av = *(const float2*)(xptr + k);    // 8B aligned (k,2*half even)
        v2f a; a[0] = av.x; a[1] = av.y;
        v2f b;
        b[0] = wptr[(long long)k * DIM];
        b[1] = wptr[(long long)k * DIM + DIM];
        // 8 args: (neg_a, A, neg_b, B, c_mod, C, reuse_a, reuse_b)
        acc = __builtin_amdgcn_wmma_f32_16x16x4_f32(
            false, a, false, b, (short)0, acc, false, false);
    }

    const float bv = bias[col];
#pragma unroll
    for (int j = 0; j < 8; ++j) {
        const int row = row_base + j + 8 * half;
        if (row < N) Y[(long long)row * DIM + col] = acc[j] + bv;
    }
}

// ---------------------------------------------------------------------------
// Per-relation in-degree (float counts; exact up to 2^24).
// ---------------------------------------------------------------------------
__global__ void degree_kernel(const int* __restrict__ dst,
                              float* __restrict__ deg, int E) {
    int e = blockIdx.x * blockDim.x + threadIdx.x;
    if (e < E) {
        __hip_atomic_fetch_add(&deg[dst[e]], 1.0f,
                               __ATOMIC_RELAXED, __HIP_MEMORY_SCOPE_AGENT);
    }
}

// ---------------------------------------------------------------------------
// Edge scatter: out[dst] += Wh[src] / deg[dst].
// One wave per edge; each lane moves a float4 (32 lanes x 4 = 128 floats).
// Destination tables are L2-resident (<=25 MB vs 192 MB L2), so the f32
// atomics resolve in L2's atomic units.
// ---------------------------------------------------------------------------
__global__ __launch_bounds__(256) void scatter_mean_kernel(
    const float* __restrict__ Wh, const int* __restrict__ src,
    const int* __restrict__ dst, const float* __restrict__ deg,
    float* __restrict__ out, int E) {
    const unsigned tid = blockIdx.x * blockDim.x + threadIdx.x;
    const int e    = (int)(tid >> 5);
    const int lane = (int)(tid & 31);
    if (e >= E) return;
    const int s  = src[e];
    const int dd = dst[e];
    const float inv = 1.0f / deg[dd];    // deg >= 1: this edge contributes
    const float4 v = *(const float4*)(Wh + (long long)s * DIM + lane * 4);
    float* o = out + (long long)dd * DIM + lane * 4;
    __hip_atomic_fetch_add(o + 0, v.x * inv, __ATOMIC_RELAXED, __HIP_MEMORY_SCOPE_AGENT);
    __hip_atomic_fetch_add(o + 1, v.y * inv, __ATOMIC_RELAXED, __HIP_MEMORY_SCOPE_AGENT);
    __hip_atomic_fetch_add(o + 2, v.z * inv, __ATOMIC_RELAXED, __HIP_MEMORY_SCOPE_AGENT);
    __hip_atomic_fetch_add(o + 3, v.w * inv, __ATOMIC_RELAXED, __HIP_MEMORY_SCOPE_AGENT);
}

// ---------------------------------------------------------------------------
extern "C" void kernel_launch(void* const* d_in, const int* in_sizes, int n_in,
                              void* d_out, int out_size, void* d_ws, size_t ws_size,
                              hipStream_t stream) {
    const float* x_paper   = (const float*)d_in[0];
    const float* x_author  = (const float*)d_in[1];
    const float* x_subject = (const float*)d_in[2];
    const float* W_wb = (const float*)d_in[3];  const float* b_wb = (const float*)d_in[4];
    const float* W_w  = (const float*)d_in[5];  const float* b_w  = (const float*)d_in[6];
    const float* W_c  = (const float*)d_in[7];  const float* b_c  = (const float*)d_in[8];
    const float* W_cd = (const float*)d_in[9];  const float* b_cd = (const float*)d_in[10];
    const float* W_ia = (const float*)d_in[11]; const float* b_ia = (const float*)d_in[12];
    const float* W_h  = (const float*)d_in[13]; const float* b_h  = (const float*)d_in[14];
    const int* src_wb = (const int*)d_in[15]; const int* dst_wb = (const int*)d_in[16];
    const int* src_w  = (const int*)d_in[17]; const int* dst_w  = (const int*)d_in[18];
    const int* src_c  = (const int*)d_in[19]; const int* dst_c  = (const int*)d_in[20];
    const int* src_cd = (const int*)d_in[21]; const int* dst_cd = (const int*)d_in[22];
    const int* src_ia = (const int*)d_in[23]; const int* dst_ia = (const int*)d_in[24];
    const int* src_h  = (const int*)d_in[25]; const int* dst_h  = (const int*)d_in[26];
    const int E_wb = in_sizes[15], E_w = in_sizes[17], E_c = in_sizes[19];
    const int E_cd = in_sizes[21], E_ia = in_sizes[23], E_h = in_sizes[25];

    // Workspace layout (floats): projected features, then contiguous degree arrays.
    float* ws = (float*)d_ws;
    size_t off = 0;
    float* Wh_wb = ws + off; off += (size_t)NPAPER   * DIM;
    float* Wh_w  = ws + off; off += (size_t)NAUTHOR  * DIM;
    float* Wh_c  = ws + off; off += (size_t)NPAPER   * DIM;
    float* Wh_cd = ws + off; off += (size_t)NPAPER   * DIM;
    float* Wh_ia = ws + off; off += (size_t)NPAPER   * DIM;
    float* Wh_h  = ws + off; off += (size_t)NSUBJECT * DIM;
    float* deg_wb = ws + off; off += NAUTHOR;   // degrees kept contiguous below
    float* deg_w  = ws + off; off += NPAPER;
    float* deg_c  = ws + off; off += NPAPER;
    float* deg_cd = ws + off; off += NPAPER;
    float* deg_h  = ws + off; off += NPAPER;
    float* deg_ia = ws + off; off += NSUBJECT;

    float* out_paper   = (float*)d_out;                         // [50000,128]
    float* out_author  = out_paper  + (size_t)NPAPER  * DIM;    // [30000,128]
    float* out_subject = out_author + (size_t)NAUTHOR * DIM;    // [60,128]

    // 1) Zero output + degree arrays (running sums / counts).
    const long long n_out = (long long)out_size;
    fill_zero_kernel<<<(unsigned)((n_out + 255) / 256), 256, 0, stream>>>((float*)d_out, n_out);
    const long long n_deg = (long long)NAUTHOR + 4LL * NPAPER + NSUBJECT;
    fill_zero_kernel<<<(unsigned)((n_deg + 255) / 256), 256, 0, stream>>>(deg_wb, n_deg);

    // 2) Per-relation linear projection at source nodes (WMMA fp32 GEMM).
    #define GEMM(X, W, B, Y, N) \
        gemm_bias_wmma_kernel<<<((N) + 15) / 16, 256, 0, stream>>>((X), (W), (B), (Y), (N))
    GEMM(x_paper,   W_wb, b_wb, Wh_wb, NPAPER);
    GEMM(x_author,  W_w,  b_w,  Wh_w,  NAUTHOR);
    GEMM(x_paper,   W_c,  b_c,  Wh_c,  NPAPER);
    GEMM(x_paper,   W_cd, b_cd, Wh_cd, NPAPER);
    GEMM(x_paper,   W_ia, b_ia, Wh_ia, NPAPER);
    GEMM(x_subject, W_h,  b_h,  Wh_h,  NSUBJECT);
    #undef GEMM

    // 3) Degrees per relation.
    #define DEG(DST, DEGP, E) \
        degree_kernel<<<((E) + 255) / 256, 256, 0, stream>>>((DST), (DEGP), (E))
    DEG(dst_wb, deg_wb, E_wb);
    DEG(dst_w,  deg_w,  E_w);
    DEG(dst_c,  deg_c,  E_c);
    DEG(dst_cd, deg_cd, E_cd);
    DEG(dst_h,  deg_h,  E_h);
    DEG(dst_ia, deg_ia, E_ia);
    #undef DEG

    // 4) Edge scatter with fused mean (sum of per-relation means lands in out).
    #define SCAT(WH, SRC, DST, DEGP, OUT, E) \
        scatter_mean_kernel<<<((E) + 7) / 8, 256, 0, stream>>>((WH), (SRC), (DST), (DEGP), (OUT), (E))
    SCAT(Wh_wb, src_wb, dst_wb, deg_wb, out_author,  E_wb);
    SCAT(Wh_w,  src_w,  dst_w,  deg_w,  out_paper,   E_w);
    SCAT(Wh_c,  src_c,  dst_c,  deg_c,  out_paper,   E_c);
    SCAT(Wh_cd, src_cd, dst_cd, deg_cd, out_paper,   E_cd);
    SCAT(Wh_h,  src_h,  dst_h,  deg_h,  out_paper,   E_h);
    SCAT(Wh_ia, src_ia, dst_ia, deg_ia, out_subject, E_ia);
    #undef SCAT

    (void)n_in; (void)ws_size;
}